// GCNRelationModel_29300266893658
// MI455X (gfx1250) — compile-verified
//
#include <hip/hip_runtime.h>
#include <hip/hip_bf16.h>

// ---------------------------------------------------------------------------
// GCNRelationModel forward for MI455X (gfx1250), all-f32.
//   1. k_init_out : d_out[0:8192] = b_mlp, d_out[8192:17792] = 0
//   2. k_relconv  : per-sample WG, 300x256 f32 accumulator in LDS (300KB --
//                   needs CDNA5's 320KB/WGP LDS), ds_add_f32 scatter,
//                   TENSOR_STORE_FROM_LDS (TDM) writeout
//   3. k_gemm_bias x3 : Q/K/V = h @ W + b   (f32 WMMA 16x16x4)
//   4. k_attn     : per-(b,head) WG; K/V staged into LDS by two
//                   TENSOR_LOAD_TO_LDS 2D descriptors (strided 300x32 tile);
//                   online softmax + head/query-averaged score sums
//   5. k_gemm_bias: outs = ctx @ Wo + bo    (f32 WMMA)
//   6. k_mlp      : gcn = outs @ W_mlp, split-K, f32 WMMA + global f32 atomics
// ---------------------------------------------------------------------------

typedef float    v2f __attribute__((ext_vector_type(2)));
typedef float    v8f __attribute__((ext_vector_type(8)));
typedef unsigned v4u __attribute__((ext_vector_type(4)));
typedef int      v4i __attribute__((ext_vector_type(4)));
typedef int      v8i __attribute__((ext_vector_type(8)));

#define NN      300                 // N_NODES
#define NE      9000                // N_EDGES
#define HID     256
#define NHEAD   8
#define HDIM    32
#define NBLK    128                 // bdd blocks of 2x2
#define BATCH   32
#define ROWS    (BATCH * NN)        // 9600
#define ETOT    (NE + NN)           // 9300 (incl self loops)
#define KMLP    (NN * HID)          // 76800
#define KCHUNK  960                 // split-K chunk: 80 chunks, 240 wmma each
#define OUT_GCN (BATCH * HID)       // 8192
#define OUT_ALL (OUT_GCN + BATCH * NN)  // 17792

// ------------------------- TDM descriptor helpers --------------------------
// D# layout per CDNA5 ISA. 2D tile of 4-byte elements:
//   group0: count=1 | lds_addr | global_addr[56:0] | type=2
//   group1: data_size=2(4B) | tensor_dim0/1 | tile_dim0/1 | dim0 stride
// Groups 2/3 zero (2D). Issued by one wave per workgroup (TDM ignores EXEC).
__device__ __forceinline__ unsigned lds_off32(const void* p) {
  // generic pointer to LDS: ADDR[31:0] is the LDS byte offset (ISA 10.2)
  return (unsigned)(unsigned long long)p;
}

__device__ __forceinline__ void tdm_make_desc(v4u& g0, v8i& g1,
    unsigned lds_addr, const void* gptr,
    unsigned tens_d0, unsigned tens_d1,
    unsigned tile_d0, unsigned tile_d1, unsigned d0_stride) {
  unsigned long long ga = (unsigned long long)gptr;
  g0 = (v4u){1u,                                   // count=1, user-mode load/store
             lds_addr,                             // lds_addr (bytes)
             (unsigned)ga,                         // global_addr[31:0]
             ((unsigned)(ga >> 32) & 0x01FFFFFFu)  // global_addr[56:32]
                 | 0x80000000u};                   // type=2 (bits 127:126 = 10b)
  g1 = (v8i){(int)(2u << 16),                            // data_size=2 -> 4 bytes
             (int)((tens_d0 & 0xFFFFu) << 16),           // tensor_dim0[15:0]
             (int)((tens_d0 >> 16) | ((tens_d1 & 0xFFFFu) << 16)),
             (int)((tens_d1 >> 16) | (tile_d0 << 16)),   // tile_dim0
             (int)(tile_d1 & 0xFFFFu),                   // tile_dim1 (tile_dim2=0)
             (int)d0_stride,                             // tensor_dim0_stride[31:0]
             0, 0};
}

__device__ __forceinline__ void tdm_load_2d(unsigned lds_addr, const void* gptr,
    unsigned tens_d0, unsigned tens_d1, unsigned tile_d0, unsigned tile_d1,
    unsigned d0_stride) {
  v4u g0; v8i g1;
  tdm_make_desc(g0, g1, lds_addr, gptr, tens_d0, tens_d1, tile_d0, tile_d1, d0_stride);
  v4i gz = (v4i){0, 0, 0, 0};
#if __clang_major__ >= 23
  v8i gz8 = (v8i){0, 0, 0, 0, 0, 0, 0, 0};
  __builtin_amdgcn_tensor_load_to_lds(g0, g1, gz, gz, gz8, 0);
#else
  __builtin_amdgcn_tensor_load_to_lds(g0, g1, gz, gz, 0);
#endif
}

__device__ __forceinline__ void tdm_store_2d(unsigned lds_addr, void* gptr,
    unsigned tens_d0, unsigned tens_d1, unsigned tile_d0, unsigned tile_d1,
    unsigned d0_stride) {
  v4u g0; v8i g1;
  tdm_make_desc(g0, g1, lds_addr, gptr, tens_d0, tens_d1, tile_d0, tile_d1, d0_stride);
  v4i gz = (v4i){0, 0, 0, 0};
#if __clang_major__ >= 23
  v8i gz8 = (v8i){0, 0, 0, 0, 0, 0, 0, 0};
  __builtin_amdgcn_tensor_store_from_lds(g0, g1, gz, gz, gz8, 0);
#else
  __builtin_amdgcn_tensor_store_from_lds(g0, g1, gz, gz, 0);
#endif
}

// --------------------------- output init ----------------------------------
__global__ void k_init_out(float* __restrict__ out, const float* __restrict__ b_mlp) {
  int i = blockIdx.x * blockDim.x + threadIdx.x;
  if (i < OUT_GCN)      out[i] = b_mlp[i & (HID - 1)];
  else if (i < OUT_ALL) out[i] = 0.0f;
}

// ------------------- RelGraphConv (bdd) with LDS segment-sum ---------------
// One workgroup per sample; LDS holds the full 300x256 f32 output (307200 B).
// Writeout via one TDM tensor_store_from_lds descriptor (wave 0).
__global__ void k_relconv(const float* __restrict__ fea,
                          const int* __restrict__ esrc, const int* __restrict__ edst,
                          const int* __restrict__ etyp, const int* __restrict__ self_et,
                          const float* __restrict__ W_rel, const float* __restrict__ b_rel,
                          float* __restrict__ h) {
  extern __shared__ float hl[];               // [NN*HID]
  const int b   = blockIdx.x;
  const int tid = threadIdx.x;
  for (int i = tid; i < NN * HID; i += blockDim.x) hl[i] = b_rel[i & (HID - 1)];
  __syncthreads();

  const float* feab = fea + (size_t)b * NN * HID;
  // task = (edge, block): y[2] = x[2] @ W_et[blk]  (einsum 'ebi,ebio->ebo')
  for (int t = tid; t < ETOT * NBLK; t += blockDim.x) {
    const int e = t >> 7, blk = t & (NBLK - 1);
    int src, dst, et;
    if (e < NE) {
      src = esrc[b * NE + e]; dst = edst[b * NE + e]; et = etyp[b * NE + e];
    } else {
      src = dst = e - NE;     et = self_et[b * NN + src];
    }
    const float x0 = feab[src * HID + 2 * blk];
    const float x1 = feab[src * HID + 2 * blk + 1];
    const float* W = W_rel + (((size_t)et * NBLK + blk) << 2);  // [i][o] 2x2
    const float y0 = x0 * W[0] + x1 * W[2];
    const float y1 = x0 * W[1] + x1 * W[3];
    atomicAdd(&hl[dst * HID + 2 * blk],     y0);   // ds_add_f32
    atomicAdd(&hl[dst * HID + 2 * blk + 1], y1);
  }
  __syncthreads();
  if (tid < 32) {                 // wave 0 issues a single 300x256 tile store
    tdm_store_2d(lds_off32(hl), h + (size_t)b * NN * HID,
                 HID, ROWS, HID, NN, HID);
    __builtin_amdgcn_s_wait_tensorcnt(0);
  }
}

// --------------- f32 WMMA GEMM: C[M,256] = A[M,256] @ W + bias -------------
// One wave per 16x16 C tile. A 16x4 layout per ISA 7.12.2:
//   lanes 0-15:  M=lane,    K = k0+{0,1} in vgpr{0,1}
//   lanes 16-31: M=lane-16, K = k0+{2,3}
// B 4x16 assumed symmetric (lane<->N, vgpr<->K)  [compile-only assumption].
__global__ void k_gemm_bias(const float* __restrict__ A, const float* __restrict__ W,
                            const float* __restrict__ bias, float* __restrict__ C) {
  const int m0 = blockIdx.x * 16, n0 = blockIdx.y * 16;
  const int lane = threadIdx.x & 31;
  const int r = lane & 15, hi = lane >> 4;
  v8f acc = {};
  const float* arow = A + (size_t)(m0 + r) * HID;
  for (int k0 = 0; k0 < HID; k0 += 4) {
    const int ka = k0 + hi * 2;
    v2f a = *(const v2f*)(arow + ka);
    v2f bb;
    bb.x = W[(size_t)ka * HID + n0 + r];
    bb.y = W[(size_t)(ka + 1) * HID + n0 + r];
    acc = __builtin_amdgcn_wmma_f32_16x16x4_f32(false, a, false, bb,
                                                (short)0, acc, false, false);
  }
  const int   col = n0 + r;
  const float bv  = bias[col];
  #pragma unroll
  for (int i = 0; i < 8; ++i) {                 // C vgpr i: row = i (+8 for hi)
    const int row = m0 + i + hi * 8;
    C[(size_t)row * HID + col] = acc[i] + bv;
  }
}

// --------------------------- attention ------------------------------------
// One workgroup per (batch, head). K/V column slices (300x32 of [9600,256])
// staged into LDS by the Tensor Data Mover: tile_dim0=32, tile_dim1=300,
// row stride=256 -- one descriptor per matrix instead of ~19k scalar loads.
__global__ void k_attn(const float* __restrict__ Q, const float* __restrict__ Kb,
                       const float* __restrict__ Vb, float* __restrict__ ctx,
                       float* __restrict__ attn_out) {
  __shared__ float Ksh[NN * HDIM];
  __shared__ float Vsh[NN * HDIM];
  __shared__ float asum[NN];
  const int b = blockIdx.x >> 3, hd = blockIdx.x & 7;
  const int tid = threadIdx.x;

  if (tid < 32) {                               // wave 0 issues both DMAs
    const size_t base = (size_t)(b * NN) * HID + hd * HDIM;
    tdm_load_2d(lds_off32(Ksh), Kb + base, HID, ROWS, HDIM, NN, HID);
    tdm_load_2d(lds_off32(Vsh), Vb + base, HID, ROWS, HDIM, NN, HID);
    __builtin_amdgcn_s_wait_tensorcnt(0);
  }
  for (int i = tid; i < NN; i += blockDim.x) asum[i] = 0.0f;
  __syncthreads();

  const float scale = 0.17677669529663687f;     // 1/sqrt(32)
  const int q = tid;
  if (q < NN) {
    const size_t qoff = (size_t)(b * NN + q) * HID + hd * HDIM;
    float qr[HDIM], ca[HDIM];
    #pragma unroll
    for (int d = 0; d < HDIM; ++d) { qr[d] = Q[qoff + d]; ca[d] = 0.0f; }
    float m = -3.0e38f, l = 0.0f;
    for (int k = 0; k < NN; ++k) {
      float s = 0.0f;
      #pragma unroll
      for (int d = 0; d < HDIM; ++d) s = fmaf(qr[d], Ksh[k * HDIM + d], s);
      s *= scale;
      const float mn   = fmaxf(m, s);
      const float corr = __expf(m - mn);
      const float p    = __expf(s - mn);
      l = l * corr + p;
      #pragma unroll
      for (int d = 0; d < HDIM; ++d)
        ca[d] = fmaf(p, Vsh[k * HDIM + d], ca[d] * corr);
      m = mn;
    }
    const float invl = 1.0f / l;
    #pragma unroll
    for (int d = 0; d < HDIM; ++d) ctx[qoff + d] = ca[d] * invl;
    // pass 2: column sums of softmax probs (staggered to avoid LDS hot-spot)
    for (int kk = 0; kk < NN; ++kk) {
      int k = q + kk; if (k >= NN) k -= NN;
      float s = 0.0f;
      #pragma unroll
      for (int d = 0; d < HDIM; ++d) s = fmaf(qr[d], Ksh[k * HDIM + d], s);
      atomicAdd(&asum[k], __expf(s * scale - m) * invl);
    }
  }
  __syncthreads();
  for (int i = tid; i < NN; i += blockDim.x)
    unsafeAtomicAdd(&attn_out[b * NN + i], asum[i] * (1.0f / (NHEAD * NN)));
}

// ------------- MLP: out[32,256] += outs[32,76800] @ W_mlp (split-K) --------
__global__ void k_mlp(const float* __restrict__ A, const float* __restrict__ Wm,
                      float* __restrict__ out) {
  const int m0 = blockIdx.x * 16, n0 = blockIdx.y * 16;
  const int kbeg = blockIdx.z * KCHUNK;
  const int lane = threadIdx.x & 31;
  const int r = lane & 15, hi = lane >> 4;
  v8f acc = {};
  const float* arow = A + (size_t)(m0 + r) * KMLP;
  for (int k0 = kbeg; k0 < kbeg + KCHUNK; k0 += 4) {
    const int ka = k0 + hi * 2;
    // stream-prefetch W_mlp 16 rows ahead (the 78.6MB bandwidth driver)
    __builtin_prefetch((const void*)(Wm + (size_t)(ka + 16) * HID + n0 + r), 0, 0);
    v2f a = *(const v2f*)(arow + ka);
    v2f bb;
    bb.x = Wm[(size_t)ka * HID + n0 + r];
    bb.y = Wm[(size_t)(ka + 1) * HID + n0 + r];
    acc = __builtin_amdgcn_wmma_f32_16x16x4_f32(false, a, false, bb,
                                                (short)0, acc, false, false);
  }
  const int col = n0 + r;
  #pragma unroll
  for (int i = 0; i < 8; ++i) {
    const int row = m0 + i + hi * 8;
    unsafeAtomicAdd(&out[(size_t)row * HID + col], acc[i]);  // global_atomic_add_f32
  }
}

// ---------------------------------------------------------------------------
extern "C" void kernel_launch(void* const* d_in, const int* in_sizes, int n_in,
                              void* d_out, int out_size, void* d_ws, size_t ws_size,
                              hipStream_t stream) {
  const float* fea     = (const float*)d_in[0];
  const int*   esrc    = (const int*)  d_in[1];
  const int*   edst    = (const int*)  d_in[2];
  const int*   etyp    = (const int*)  d_in[3];
  const int*   self_et = (const int*)  d_in[4];
  const float* W_rel   = (const float*)d_in[5];
  const float* b_rel   = (const float*)d_in[6];
  const float* Wq      = (const float*)d_in[7];
  const float* Wk      = (const float*)d_in[8];
  const float* Wv      = (const float*)d_in[9];
  const float* Wo      = (const float*)d_in[10];
  const float* bq      = (const float*)d_in[11];
  const float* bk      = (const float*)d_in[12];
  const float* bv      = (const float*)d_in[13];
  const float* bo      = (const float*)d_in[14];
  const float* W_mlp   = (const float*)d_in[15];
  const float* b_mlp   = (const float*)d_in[16];
  float* out = (float*)d_out;

  float* ws = (float*)d_ws;
  const size_t SLOT = (size_t)ROWS * HID;      // 2,457,600 floats / slot
  float* hbuf   = ws;                          // slot0: h, later ctx
  float* qbuf   = ws + SLOT;                   // slot1: q, later outs
  float* kbuf   = ws + 2 * SLOT;               // slot2: k
  float* vbuf   = ws + 3 * SLOT;               // slot3: v
  float* ctxbuf  = hbuf;                       // h dead once Q/K/V exist
  float* outsbuf = qbuf;                       // q dead once attention done

  k_init_out<<<(OUT_ALL + 255) / 256, 256, 0, stream>>>(out, b_mlp);

  k_relconv<<<BATCH, 512, NN * HID * sizeof(float), stream>>>(
      fea, esrc, edst, etyp, self_et, W_rel, b_rel, hbuf);

  dim3 gg(ROWS / 16, HID / 16);                // 600 x 16 tiles, 1 wave each
  k_gemm_bias<<<gg, 32, 0, stream>>>(hbuf, Wq, bq, qbuf);
  k_gemm_bias<<<gg, 32, 0, stream>>>(hbuf, Wk, bk, kbuf);
  k_gemm_bias<<<gg, 32, 0, stream>>>(hbuf, Wv, bv, vbuf);

  k_attn<<<BATCH * NHEAD, 320, 0, stream>>>(qbuf, kbuf, vbuf, ctxbuf, out + OUT_GCN);

  k_gemm_bias<<<gg, 32, 0, stream>>>(ctxbuf, Wo, bo, outsbuf);

  dim3 gm(BATCH / 16, HID / 16, KMLP / KCHUNK);   // 2 x 16 x 80
  k_mlp<<<gm, 32, 0, stream>>>(outsbuf, W_mlp, out);
}